// CausalSimulationModule_70463233458696
// MI455X (gfx1250) — compile-verified
//
#include <hip/hip_runtime.h>
#include <hip/hip_bf16.h>
#include <cstdint>

// ---------------------------------------------------------------------------
// CausalSimulationModule on MI455X (gfx1250), wave32.
// B=4, N=512, D=128.
// Rewrite: upd[b,i,:] = ( sum_j A[i,j] * relu(s[b,j,:]+t[b,i,:]+b1) ) @ W2 + b2
// (A rows sum to 1, W2 is j-independent) -> no (B,N,N,D) intermediate.
// GEMMs: V_WMMA_F32_16X16X4_F32 (full f32). Weight/attention tiles staged
// into LDS via the Tensor Data Mover (tensor_load_to_lds + s_wait_tensorcnt).
// ---------------------------------------------------------------------------

#define BB 4
#define NN 512
#define DD 128
#define TAU_INV 2.0f
#define EPSV 1e-9f

typedef __attribute__((ext_vector_type(2))) float v2f;
typedef __attribute__((ext_vector_type(8))) float v8f;
typedef __attribute__((ext_vector_type(4))) unsigned int v4u;
typedef __attribute__((ext_vector_type(8))) int v8i;
typedef __attribute__((ext_vector_type(4))) int v4i;

// ---------------------------------------------------------------------------
// TDM: DMA a 2D tile (tile1 rows x tile0 elems, f32) from a row-major global
// tensor (dim1 x dim0, row stride stride0 elems) into LDS at lds_off.
// Descriptor per CDNA5 ISA D# spec (8.3/8.4): group0 = control/addresses,
// group1 = dims/tile/strides, groups 2-3 zero (2D tile).
// Call from ONE wave with full EXEC; follow with s_wait_tensorcnt + barrier.
// ---------------------------------------------------------------------------
__device__ __forceinline__ void tdm_load_2d_f32(const void* gptr, uint32_t lds_off,
                                                uint32_t dim0, uint32_t dim1,
                                                uint32_t tile0, uint32_t tile1,
                                                uint32_t stride0) {
    const uint64_t ga = (uint64_t)(uintptr_t)gptr;
    v4u g0;
    g0.x = 1u;                                   // count=1 (valid user descriptor)
    g0.y = lds_off;                              // lds_addr (bytes)
    g0.z = (uint32_t)ga;                         // global_addr[31:0]
    g0.w = (uint32_t)((ga >> 32) & 0x1FFFFFFu)   // global_addr[56:32]
           | (2u << 30);                         // type=2 ("image")
    v8i g1;
    g1[0] = (int)(2u << 16);                     // data_size=2 -> 4 bytes
    g1[1] = (int)(dim0 << 16);                   // tensor_dim0[15:0]  @bits63:48
    g1[2] = (int)((dim0 >> 16) | (dim1 << 16));  // dim0[31:16] | dim1[15:0]
    g1[3] = (int)((dim1 >> 16) | (tile0 << 16)); // dim1[31:16] | tile_dim0
    g1[4] = (int)(tile1 & 0xFFFFu);              // tile_dim1 (tile_dim2=0)
    g1[5] = (int)stride0;                        // tensor_dim0_stride[31:0]
    g1[6] = 0;                                   // stride0[47:32] | dim1_stride lo
    g1[7] = 0;
    v4i z4 = {};
    v8i z8 = {};
    // 6-arg form (clang-23 / therock-10.0 headers): groups 2,3 + extra group.
    __builtin_amdgcn_tensor_load_to_lds(g0, g1, z4, z4, z8, 0);
}

// ---------------------------------------------------------------------------
// Kernel 1: A = softmax((edge_logits + gumbel(u))/TAU) row-wise.
// One block (512 threads = 16 waves) per row. Writes A workspace + the B
// broadcast copies straight into the output tensor.
// ---------------------------------------------------------------------------
__global__ void gumbel_softmax_kernel(const float* __restrict__ logits,
                                      const float* __restrict__ u,
                                      float* __restrict__ A_ws,
                                      float* __restrict__ A_out) {
    const int i = blockIdx.x;
    const int j = threadIdx.x;
    __shared__ float red[16];

    float uv = u[i * NN + j];
    float g  = -logf(-logf(uv + EPSV) + EPSV);
    float z  = (logits[i * NN + j] + g) * TAU_INV;

    float m = z;
    #pragma unroll
    for (int off = 16; off; off >>= 1) m = fmaxf(m, __shfl_xor(m, off, 32));
    if ((j & 31) == 0) red[j >> 5] = m;
    __syncthreads();
    if (j < 16) {
        float t = red[j];
        #pragma unroll
        for (int off = 8; off; off >>= 1) t = fmaxf(t, __shfl_xor(t, off, 16));
        if (j == 0) red[0] = t;
    }
    __syncthreads();
    m = red[0];
    __syncthreads();

    float e = expf(z - m);
    float s = e;
    #pragma unroll
    for (int off = 16; off; off >>= 1) s += __shfl_xor(s, off, 32);
    if ((j & 31) == 0) red[j >> 5] = s;
    __syncthreads();
    if (j < 16) {
        float t = red[j];
        #pragma unroll
        for (int off = 8; off; off >>= 1) t += __shfl_xor(t, off, 16);
        if (j == 0) red[0] = t;
    }
    __syncthreads();

    float p = e / red[0];
    A_ws[i * NN + j] = p;
    #pragma unroll
    for (int b = 0; b < BB; ++b)
        A_out[(size_t)b * NN * NN + i * NN + j] = p;
}

// ---------------------------------------------------------------------------
// Kernel 2 (generic): out[M,128] = A[M,128] @ Bm[128,128] (+bias) (+resid).
// Whole 64KB weight matrix TDM'd into LDS once per block; one 16x16 output
// tile per wave (8 waves/block) via V_WMMA_F32_16X16X4_F32, K unrolled.
// A-frag: lane<16 row l K={k,k+1}; lane>=16 K={k+2,k+3}. B-frag symmetric.
// D: VGPR r -> row = r + 8*(lane>=16), col = lane&15.
// ---------------------------------------------------------------------------
__global__ void gemm_wmma_f32_kernel(const float* __restrict__ A,
                                     const float* __restrict__ Bm,
                                     const float* __restrict__ bias,
                                     const float* __restrict__ resid,
                                     float* __restrict__ out) {
    __shared__ float ldsB[DD * DD];              // 64 KB

    if (threadIdx.x < 32) {                      // wave 0, full EXEC
        tdm_load_2d_f32(Bm, (uint32_t)(uintptr_t)&ldsB[0],
                        DD, DD, DD, DD, DD);
        __builtin_amdgcn_s_wait_tensorcnt(0);
    }
    __syncthreads();

    const int wave = threadIdx.x >> 5;           // 0..7
    const int lane = threadIdx.x & 31;
    const int tile = blockIdx.x * 8 + wave;      // 1024 tiles total
    const int row0 = (tile >> 3) << 4;
    const int col0 = (tile & 7) << 4;
    const int half = lane >> 4;
    const int l    = lane & 15;
    const int colL = col0 + l;

    const float* Ab = A + (size_t)row0 * DD;

    v8f acc = {};
    #pragma unroll
    for (int k = 0; k < DD; k += 4) {
        const int ka = k + half * 2;
        v2f av, bv;
        av.x = Ab[l * DD + ka];
        av.y = Ab[l * DD + ka + 1];
        bv.x = ldsB[ka * DD + colL];
        bv.y = ldsB[(ka + 1) * DD + colL];
        acc = __builtin_amdgcn_wmma_f32_16x16x4_f32(
            false, av, false, bv, (short)0, acc, false, false);
    }

    const float badd = bias ? bias[colL] : 0.0f;
    #pragma unroll
    for (int r = 0; r < 8; ++r) {
        const int row = row0 + r + half * 8;
        float x = acc[r] + badd;
        if (resid) x += resid[(size_t)row * DD + colL];
        out[(size_t)row * DD + colL] = x;
    }
}

// ---------------------------------------------------------------------------
// Kernel 3: v[b,i,k] = sum_j A[i,j] * relu(s[b,j,k] + c[b,i,k]).
// Block = 128 threads (k), TI=8 i-values per block; the 8 A-rows (16 KB) are
// TDM'd into LDS so each streamed s[b,j,:] load is reused 8x. grid=(N/TI, B).
// ---------------------------------------------------------------------------
#define TI 8
__global__ void gather_relu_kernel(const float* __restrict__ s,
                                   const float* __restrict__ c,
                                   const float* __restrict__ A_ws,
                                   float* __restrict__ v) {
    const int b  = blockIdx.y;
    const int i0 = blockIdx.x * TI;
    const int k  = threadIdx.x;                  // 0..127

    __shared__ float Arow[TI][NN];               // 16 KB

    if (threadIdx.x < 32) {                      // wave 0, full EXEC
        tdm_load_2d_f32(A_ws + (size_t)i0 * NN,
                        (uint32_t)(uintptr_t)&Arow[0][0],
                        NN, NN, NN, TI, NN);
        __builtin_amdgcn_s_wait_tensorcnt(0);
    }
    __syncthreads();

    float cc[TI], acc[TI];
    #pragma unroll
    for (int t = 0; t < TI; ++t) {
        cc[t]  = c[((size_t)b * NN + (i0 + t)) * DD + k];
        acc[t] = 0.0f;
    }

    const float* sb = s + (size_t)b * NN * DD;
    for (int j = 0; j < NN; ++j) {
        const float sv = sb[(size_t)j * DD + k];
        #pragma unroll
        for (int t = 0; t < TI; ++t) {
            const float h = fmaxf(sv + cc[t], 0.0f);
            acc[t] = fmaf(Arow[t][j], h, acc[t]);
        }
    }

    #pragma unroll
    for (int t = 0; t < TI; ++t)
        v[((size_t)b * NN + (i0 + t)) * DD + k] = acc[t];
}

// ---------------------------------------------------------------------------
// Host-side launch. ws layout (floats):
//   proj[0 .. 262144) | s[.. 524288) | c[.. 786432) | v[.. 1048576) | A[.. 1310720)
// d_out layout: next_state (B*N*D) then A broadcast (B*N*N).
// ---------------------------------------------------------------------------
extern "C" void kernel_launch(void* const* d_in, const int* in_sizes, int n_in,
                              void* d_out, int out_size, void* d_ws, size_t ws_size,
                              hipStream_t stream) {
    const float* ES   = (const float*)d_in[0];   // (B,N,D)
    const float* Wp   = (const float*)d_in[1];   // (D,D)
    const float* bp   = (const float*)d_in[2];   // (D,)
    const float* EL   = (const float*)d_in[3];   // (N,N)
    const float* W1   = (const float*)d_in[4];   // (2D,D)
    const float* b1   = (const float*)d_in[5];   // (D,)
    const float* W2   = (const float*)d_in[6];   // (D,D)
    const float* b2   = (const float*)d_in[7];   // (D,)
    const float* UN   = (const float*)d_in[8];   // (N,N)

    float* out_next = (float*)d_out;                         // B*N*D
    float* out_A    = (float*)d_out + (size_t)BB * NN * DD;  // B*N*N

    float* ws    = (float*)d_ws;
    float* proj  = ws;
    float* sbuf  = ws + 1 * (size_t)BB * NN * DD;
    float* cbuf  = ws + 2 * (size_t)BB * NN * DD;
    float* vbuf  = ws + 3 * (size_t)BB * NN * DD;
    float* A_ws  = ws + 4 * (size_t)BB * NN * DD;            // N*N

    const float* W1_src = W1;                // rows [0,128)
    const float* W1_tgt = W1 + DD * DD;      // rows [128,256)

    // 1) A = gumbel-softmax (+ broadcast copies into output).
    gumbel_softmax_kernel<<<NN, NN, 0, stream>>>(EL, UN, A_ws, out_A);

    // 2) proj = ES @ Wp + bp
    gemm_wmma_f32_kernel<<<128, 256, 0, stream>>>(ES, Wp, bp, nullptr, proj);

    // 3) s = proj @ W1_src
    gemm_wmma_f32_kernel<<<128, 256, 0, stream>>>(proj, W1_src, nullptr, nullptr, sbuf);

    // 4) c = proj @ W1_tgt + b1
    gemm_wmma_f32_kernel<<<128, 256, 0, stream>>>(proj, W1_tgt, b1, nullptr, cbuf);

    // 5) v[b,i,:] = sum_j A[i,j] * relu(s[b,j,:] + c[b,i,:])
    gather_relu_kernel<<<dim3(NN / TI, BB), 128, 0, stream>>>(sbuf, cbuf, A_ws, vbuf);

    // 6) next_state = v @ W2 + b2 + ES
    gemm_wmma_f32_kernel<<<128, 256, 0, stream>>>(vbuf, W2, b2, ES, out_next);
}